// GNN_41832981463599
// MI455X (gfx1250) — compile-verified
//
#include <hip/hip_runtime.h>
#include <hip/hip_bf16.h>

// ---------------------------------------------------------------------------
// 2-layer GCN for MI455X (gfx1250).
//   layer: out = Dinv * (A+I) * Dinv * (X @ W) + b   (ReLU between layers)
// GEMMs use V_WMMA_F32_16X16X4_F32 (exact fp32, matches reference math).
// W is staged into LDS K-pair-interleaved so each B fragment is a single
// contiguous ds_load_b64 into an aligned VGPR pair (no v_mov repacking).
// The self-loop + bias initialization is fused into the GEMM epilogue,
// saving a full read+write pass over the feature matrix.
// Edge aggregation = wave-per-edge gather + f32 global atomics; all hot data
// (25.6MB feature matrix) is L2-resident on the 192MB L2.
// ---------------------------------------------------------------------------

typedef __attribute__((ext_vector_type(2))) float v2f;
typedef __attribute__((ext_vector_type(8))) float v8f;

// ------------------------------ degree / dinv ------------------------------

__global__ void gcn_deg_init(float* __restrict__ deg, int n) {
  int i = blockIdx.x * blockDim.x + threadIdx.x;
  if (i < n) deg[i] = 1.0f;  // self-loop
}

__global__ void gcn_deg_accum(const int* __restrict__ dst, float* __restrict__ deg, int e) {
  int i = blockIdx.x * blockDim.x + threadIdx.x;
  if (i < e) atomicAdd(&deg[dst[i]], 1.0f);
}

__global__ void gcn_dinv(const float* __restrict__ deg, float* __restrict__ dinv, int n) {
  int i = blockIdx.x * blockDim.x + threadIdx.x;
  if (i < n) dinv[i] = rsqrtf(deg[i]);
}

// ------------------------------ WMMA fp32 GEMM -----------------------------
// Out[M x COUT]  = A[M x 128] @ W[128 x COUT]          (optional ReLU on A)
// Self[M x COUT] = Out * dinv[row]^2 + bias[col]       (fused epilogue)
// Block = 128 threads (4 waves); each wave computes a 16-row slab.
// LDS W layout: pair-interleaved, sW[(k/2)*PSTR + 2*c + (k&1)], PSTR = 2*COUT+32
//   -> B frag (two K values, one column) is one aligned ds_load_b64, and the
//      two lane-halves (adjacent K-pairs) hit disjoint bank sets.
// LDS A layout: row stride 132 -> A-frag reads spread over 16 distinct banks.

template <int COUT, bool RELU>
__global__ __launch_bounds__(128) void gcn_gemm_wmma(
    const float* __restrict__ A, const float* __restrict__ W,
    const float* __restrict__ dinv, const float* __restrict__ bias,
    float* __restrict__ Out, float* __restrict__ Self, int M) {
  constexpr int K = 128;
  constexpr int WAVES = 4;
  constexpr int ASTR = 132;
  constexpr int PSTR = 2 * COUT + 32;  // floats per K-pair row
  constexpr int NT = COUT / 16;        // N tiles per wave

  __shared__ float sW[(K / 2) * PSTR];
  __shared__ float sA[WAVES * 16 * ASTR];

  const int tid  = threadIdx.x;
  const int wave = tid >> 5;
  const int lane = tid & 31;
  const int half = lane >> 4;  // 0: lanes 0-15, 1: lanes 16-31
  const int mn   = lane & 15;  // matrix row (A frag) / col (B, C frags)

  // Stage W pair-interleaved: per iteration a thread owns K-pair p, cols c..c+3.
  // Loads two row-major float4s, writes four ds_store_b64 pairs.
  for (int idx = tid; idx < (K / 2) * (COUT / 4); idx += 128) {
    const int p = idx / (COUT / 4);
    const int c = (idx - p * (COUT / 4)) * 4;
    const float4 r0 = *(const float4*)(W + (2 * p + 0) * COUT + c);
    const float4 r1 = *(const float4*)(W + (2 * p + 1) * COUT + c);
    float* dp = &sW[p * PSTR + 2 * c];
    *(v2f*)(dp + 0) = (v2f){r0.x, r1.x};
    *(v2f*)(dp + 2) = (v2f){r0.y, r1.y};
    *(v2f*)(dp + 4) = (v2f){r0.z, r1.z};
    *(v2f*)(dp + 6) = (v2f){r0.w, r1.w};
  }

  // Stage this wave's 16 x 128 A slab (row clamp keeps EXEC uniform).
  const long rowBase = (long)blockIdx.x * (WAVES * 16) + wave * 16;
  for (int r = 0; r < 16; ++r) {
    long row = rowBase + r;
    if (row >= M) row = M - 1;
    float4 v = *(const float4*)(A + row * (long)K + lane * 4);
    if (RELU) {
      v.x = fmaxf(v.x, 0.0f); v.y = fmaxf(v.y, 0.0f);
      v.z = fmaxf(v.z, 0.0f); v.w = fmaxf(v.w, 0.0f);
    }
    *(float4*)(&sA[(wave * 16 + r) * ASTR + lane * 4]) = v;
  }
  __syncthreads();

  v8f acc[NT] = {};
  const float* aBase = &sA[wave * 16 * ASTR];

  // 32 K-steps of 4; 16x4 A frag: lane half selects K pair {k0,k0+1} vs
  // {k0+2,k0+3}; both frags are single aligned b64 LDS loads.
  for (int k0 = 0; k0 < K; k0 += 4) {
    const int kk = k0 + 2 * half;         // even
    const v2f a = *(const v2f*)&aBase[mn * ASTR + kk];
    const int pRow = (kk >> 1) * PSTR;
#pragma unroll
    for (int n = 0; n < NT; ++n) {
      const v2f b = *(const v2f*)&sW[pRow + 2 * (n * 16 + mn)];
      acc[n] = __builtin_amdgcn_wmma_f32_16x16x4_f32(
          /*neg_a=*/false, a, /*neg_b=*/false, b,
          /*c_mod=*/(short)0, acc[n], /*reuse_a=*/false, /*reuse_b=*/false);
    }
  }

  // Fused epilogue. C/D layout: VGPR i -> row (i + 8*half), col = mn.
  float d2[8];
#pragma unroll
  for (int i = 0; i < 8; ++i) {
    const long row = rowBase + i + 8 * half;
    const float di = (row < M) ? dinv[row] : 0.0f;
    d2[i] = di * di;
  }
#pragma unroll
  for (int n = 0; n < NT; ++n) {
    const int col = n * 16 + mn;
    const float bc = bias[col];
#pragma unroll
    for (int i = 0; i < 8; ++i) {
      const long row = rowBase + i + 8 * half;
      if (row < M) {
        const float v = acc[n][i];
        Out[row * (long)COUT + col]  = v;
        Self[row * (long)COUT + col] = v * d2[i] + bc;
      }
    }
  }
}

// ------------------------------ edge scatter -------------------------------
// One edge per wave (32 lanes). Each lane handles C/32 contiguous channels.
// out[dst] += h[src] * dinv[src]*dinv[dst], accumulated with f32 atomics.

template <int C>
__global__ void gcn_edge_scatter(const float* __restrict__ h,
                                 const float* __restrict__ dinv,
                                 const int* __restrict__ src,
                                 const int* __restrict__ dst,
                                 float* __restrict__ out, int e) {
  const long gid = (long)blockIdx.x * blockDim.x + threadIdx.x;
  const int edge = (int)(gid >> 5);
  const int lane = (int)(gid & 31);
  if (edge >= e) return;

  const int s = src[edge];
  const int d = dst[edge];

  // Look ahead: pull a future source row toward L2/L0 (global_prefetch_b8).
  const int e2 = edge + 64;
  if (e2 < e) __builtin_prefetch(&h[(size_t)src[e2] * C], 0, 1);

  const float ns = dinv[s] * dinv[d];
  const float* hp = h + (size_t)s * C + lane * (C / 32);
  float* op = out + (size_t)d * C + lane * (C / 32);

  if constexpr (C == 128) {
    const float4 v = *(const float4*)hp;
    atomicAdd(&op[0], v.x * ns);
    atomicAdd(&op[1], v.y * ns);
    atomicAdd(&op[2], v.z * ns);
    atomicAdd(&op[3], v.w * ns);
  } else {
    const float2 v = *(const float2*)hp;
    atomicAdd(&op[0], v.x * ns);
    atomicAdd(&op[1], v.y * ns);
  }
}

// ------------------------------- launcher ----------------------------------

extern "C" void kernel_launch(void* const* d_in, const int* in_sizes, int n_in,
                              void* d_out, int out_size, void* d_ws, size_t ws_size,
                              hipStream_t stream) {
  const float* x  = (const float*)d_in[0];
  const int*   ei = (const int*)d_in[1];
  const float* W1 = (const float*)d_in[2];
  const float* b1 = (const float*)d_in[3];
  const float* W2 = (const float*)d_in[4];
  const float* b2 = (const float*)d_in[5];
  float* out = (float*)d_out;

  const int N = in_sizes[0] / 128;  // 50000
  const int E = in_sizes[1] / 2;    // 600000
  const int* src = ei;
  const int* dst = ei + E;

  // Workspace layout (floats): deg[N] dinv[N] h1[N*128] agg1[N*128] h2[N*64]
  float* ws   = (float*)d_ws;
  float* deg  = ws;
  float* dinv = deg + N;
  float* h1   = dinv + N;
  float* agg1 = h1 + (size_t)N * 128;
  float* h2   = agg1 + (size_t)N * 128;

  const int T = 256;
  const int nBlk  = (N + T - 1) / T;
  const int eBlk  = (E + T - 1) / T;
  const int esBlk = (int)(((long)E * 32 + T - 1) / T);
  const int gBlk  = (N + 63) / 64;  // 4 waves x 16 rows per block

  // Degrees (in-degree + self-loop) and Dinv.
  gcn_deg_init<<<nBlk, T, 0, stream>>>(deg, N);
  gcn_deg_accum<<<eBlk, T, 0, stream>>>(dst, deg, E);
  gcn_dinv<<<nBlk, T, 0, stream>>>(deg, dinv, N);

  // Layer 1: h1 = x @ W1, agg1 = h1*dinv^2 + b1 (fused); then edge scatter.
  gcn_gemm_wmma<128, false><<<gBlk, 128, 0, stream>>>(x, W1, dinv, b1, h1, agg1, N);
  gcn_edge_scatter<128><<<esBlk, T, 0, stream>>>(h1, dinv, src, dst, agg1, E);

  // Layer 2: h2 = relu(agg1) @ W2, out = h2*dinv^2 + b2 (fused); then scatter.
  gcn_gemm_wmma<64, true><<<gBlk, 128, 0, stream>>>(agg1, W2, dinv, b2, h2, out, N);
  gcn_edge_scatter<64><<<esBlk, T, 0, stream>>>(h2, dinv, src, dst, out, E);
}